// ParallelAccumulator_84713934946553
// MI455X (gfx1250) — compile-verified
//
#include <hip/hip_runtime.h>
#include <hip/hip_bf16.h>

// CDNA5 (gfx1250) wave32 WMMA reduction.
// out[b][bit] = sum_n x[b][n][bit],  x: (B, N, 32) fp32.
//
// D = A(16x4, all ones) x B(4x16) + C  via V_WMMA_F32_16X16X4_F32.
// B columns = 16 bit-lanes (two WMMAs for 32 bits), B K-rows = 4 N-rows.
// Because A is all-ones, any K-row permutation of B yields the same column
// sums, so per-lane load addresses can be chosen purely for coalescing.

typedef float v2f __attribute__((ext_vector_type(2)));
typedef float v8f __attribute__((ext_vector_type(8)));

#define BITS 32
#define NW   16   // waves per block (512 threads)

__global__ __launch_bounds__(NW * 32)
void ParallelAccumulator_wmma_reduce(const float* __restrict__ x,
                                     float* __restrict__ out,
                                     int N) {
    __shared__ float partial[NW][BITS];

    const int lane = threadIdx.x & 31;
    const int wave = threadIdx.x >> 5;
    const int b    = blockIdx.x;

    const int chunk = N / NW;      // N-rows per wave (4096/16 = 256)
    const int iters = chunk / 4;   // K=4 rows per WMMA iteration

    const int half = lane >> 4;    // which 16-lane half of the wave
    const int col  = lane & 15;    // B-matrix column (bit index within half)

    const float* p = x + (size_t)b * (size_t)N * BITS
                       + (size_t)wave * (size_t)chunk * BITS;

    v2f ones; ones.x = 1.0f; ones.y = 1.0f;   // A = 16x4 all-ones
    v8f acc0 = {};                             // bits  0..15 accumulator
    v8f acc1 = {};                             // bits 16..31 accumulator

    for (int i = 0; i < iters; ++i) {
        const float* q = p + (size_t)i * 4 * BITS;

        // Prefetch ~2KB (16 N-rows) ahead; streaming pattern.
        __builtin_prefetch(q + 16 * BITS, 0, 0);

        // Wave covers rows n0..n0+3, bits 0..31 => 512 contiguous bytes.
        // Lanes 0-15 and 16-31 each read 64B contiguous per load.
        v2f b_lo, b_hi;
        b_lo.x = __builtin_nontemporal_load(q + (half    ) * BITS + col);
        b_lo.y = __builtin_nontemporal_load(q + (half + 2) * BITS + col);
        b_hi.x = __builtin_nontemporal_load(q + (half    ) * BITS + 16 + col);
        b_hi.y = __builtin_nontemporal_load(q + (half + 2) * BITS + 16 + col);

        // D = ones x B + C  (exact fp32 column sums over 4 N-rows)
        acc0 = __builtin_amdgcn_wmma_f32_16x16x4_f32(
            false, ones, false, b_lo, (short)0, acc0, false, false);
        acc1 = __builtin_amdgcn_wmma_f32_16x16x4_f32(
            false, ones, false, b_hi, (short)0, acc1, false, false);
    }

    // C/D layout: VGPR0 holds M=0 (lanes 0-15) and M=8 (lanes 16-31), col =
    // lane%16. All D rows are identical (A all-ones), so every lane's acc[0]
    // is the column sum for its col. Lane l therefore holds bit l's partial:
    // lanes 0-15 from acc0 (bits 0-15), lanes 16-31 from acc1 (bits 16-31).
    float val = (lane < 16) ? acc0[0] : acc1[0];
    partial[wave][lane] = val;
    __syncthreads();

    if (threadIdx.x < BITS) {
        float s = 0.0f;
        #pragma unroll
        for (int w = 0; w < NW; ++w) s += partial[w][threadIdx.x];
        out[(size_t)b * BITS + threadIdx.x] = s;
    }
}

extern "C" void kernel_launch(void* const* d_in, const int* in_sizes, int n_in,
                              void* d_out, int out_size, void* d_ws, size_t ws_size,
                              hipStream_t stream) {
    const float* x  = (const float*)d_in[0];
    float* out      = (float*)d_out;
    // out_size = B*32 ; in_sizes[0] = B*N*32
    const int B = out_size / BITS;
    const int N = in_sizes[0] / out_size;   // 4096
    ParallelAccumulator_wmma_reduce<<<B, NW * 32, 0, stream>>>(x, out, N);
}